// Transformer_70643622084881
// MI455X (gfx1250) — compile-verified
//
#include <hip/hip_runtime.h>
#include <hip/hip_bf16.h>

// ---------------------------------------------------------------------------
// Transformer forward for MI455X (gfx1250), fp32 via V_WMMA_F32_16X16X4_F32.
// Register-blocked: one wave32 computes a 16x64 (tn) or 64x16 (nn) C block
// with 4 independent accumulator chains.  The K loop is chunked by 16 and
// DOUBLE-BUFFERED in registers: chunk i+1's 20 loads are issued before chunk
// i's 16 WMMAs execute, so VMEM latency is hidden under matrix work instead
// of s_wait_loadcnt 0 stalls.  All K dims (64/512/2048) are multiples of 32.
// A-matrix lane layout (ISA 7.12.2, 32-bit A 16x4): lane = hi*16 + r,
//   v0 = A[r][k0+2*hi], v1 = A[r][k0+2*hi+1].
// B-matrix 4x16 (mirrored):  v0 = B[k0+2*hi][r], v1 = B[k0+2*hi+1][r].
// C/D 16x16: VGPR i, lane hi*16+r -> element (row = i + 8*hi, col = r).
// ---------------------------------------------------------------------------

typedef float v2f __attribute__((ext_vector_type(2)));
typedef float v8f __attribute__((ext_vector_type(8)));

#define WPB     8      // waves per block (256 threads)
#define D_MODEL 512
#define NHEAD   8
#define DHEAD   64
#define NLAYER  6
#define FFDIM   2048
#define VOCAB   32000
#define BATCH   4
#define SEQ     512
#define EMB_SCALE 22.627416997969522f   // sqrt(512)

__device__ __forceinline__ v8f wmma4f32(v2f a, v2f b, v8f c) {
  // 8 args: (neg_a, A, neg_b, B, c_mod, C, reuse_a, reuse_b)
  return __builtin_amdgcn_wmma_f32_16x16x4_f32(false, a, false, b, (short)0, c,
                                               false, false);
}

struct ChunkTN { v2f a[4], b0[4], b1[4], b2[4], b3[4]; };

__device__ __forceinline__ void load_tn(ChunkTN& c, const float* Arow,
                                        const float* W0, const float* W1,
                                        const float* W2, const float* W3,
                                        int k0) {
#pragma unroll
  for (int kk = 0; kk < 4; ++kk) c.a[kk]  = *(const v2f*)(Arow + k0 + 4 * kk);
#pragma unroll
  for (int kk = 0; kk < 4; ++kk) c.b0[kk] = *(const v2f*)(W0 + k0 + 4 * kk);
#pragma unroll
  for (int kk = 0; kk < 4; ++kk) c.b1[kk] = *(const v2f*)(W1 + k0 + 4 * kk);
#pragma unroll
  for (int kk = 0; kk < 4; ++kk) c.b2[kk] = *(const v2f*)(W2 + k0 + 4 * kk);
#pragma unroll
  for (int kk = 0; kk < 4; ++kk) c.b3[kk] = *(const v2f*)(W3 + k0 + 4 * kk);
}

__device__ __forceinline__ void mma_tn(v8f acc[4], const ChunkTN& c) {
#pragma unroll
  for (int kk = 0; kk < 4; ++kk) {
    acc[0] = wmma4f32(c.a[kk], c.b0[kk], acc[0]);
    acc[1] = wmma4f32(c.a[kk], c.b1[kk], acc[1]);
    acc[2] = wmma4f32(c.a[kk], c.b2[kk], acc[2]);
    acc[3] = wmma4f32(c.a[kk], c.b3[kk], acc[3]);
  }
}

// C[m, n] = scale * sum_k A[m,k]*W[n,k] (+bias[n]) (+relu)
// One wave computes a 16 x 64 block: 4 accumulators, A fragment reused 4x.
// Requires N % 64 == 0, M % 16 == 0, K % 32 == 0.
__global__ __launch_bounds__(WPB * 32)
void gemm_tn_kernel(const float* __restrict__ A, const float* __restrict__ W,
                    const float* __restrict__ bias, float* __restrict__ C,
                    int M, int N, int K, int lda, int ldw, int ldc, int nh,
                    size_t aBS, size_t aHS, size_t wBS, size_t wHS,
                    size_t cBS, size_t cHS, float scale, int act) {
  const int lane = threadIdx.x & 31;
  const int wave = threadIdx.x >> 5;
  const int tilesN = N >> 6;                    // 64-wide N blocks
  const int tilesM = M >> 4;
  const int tile = blockIdx.x * WPB + wave;
  if (tile >= tilesM * tilesN) return;          // wave-uniform exit
  const int tm = tile / tilesN;
  const int tn = tile - tm * tilesN;
  const int r  = lane & 15;
  const int hi = lane >> 4;
  const int zb = blockIdx.z / nh;
  const int zh = blockIdx.z - zb * nh;

  const float* Arow = A + (size_t)zb * aBS + (size_t)zh * aHS +
                      (size_t)(tm * 16 + r) * lda + 2 * hi;
  const float* Wbase = W + (size_t)zb * wBS + (size_t)zh * wHS + 2 * hi;
  const float* W0 = Wbase + (size_t)(tn * 64 + r) * ldw;
  const float* W1 = W0 + (size_t)16 * ldw;
  const float* W2 = W0 + (size_t)32 * ldw;
  const float* W3 = W0 + (size_t)48 * ldw;

  v8f acc[4];
#pragma unroll
  for (int j = 0; j < 4; ++j)
    acc[j] = (v8f){0.f, 0.f, 0.f, 0.f, 0.f, 0.f, 0.f, 0.f};

  ChunkTN U, V;
  load_tn(U, Arow, W0, W1, W2, W3, 0);          // prologue: chunk 0
#pragma unroll 1
  for (int k0 = 0; k0 < K; k0 += 32) {
    load_tn(V, Arow, W0, W1, W2, W3, k0 + 16);  // prefetch chunk k0+16
    mma_tn(acc, U);                             // compute chunk k0
    if (k0 + 32 < K)
      load_tn(U, Arow, W0, W1, W2, W3, k0 + 32);// prefetch chunk k0+32
    mma_tn(acc, V);                             // compute chunk k0+16
  }

  float* Cb = C + (size_t)zb * cBS + (size_t)zh * cHS;
#pragma unroll
  for (int j = 0; j < 4; ++j) {
    const int col = tn * 64 + j * 16 + r;
    const float bn = bias ? bias[col] : 0.0f;
#pragma unroll
    for (int i = 0; i < 8; ++i) {
      const int row = tm * 16 + i + 8 * hi;
      float v = acc[j][i] * scale + bn;
      if (act) v = fmaxf(v, 0.0f);
      Cb[(size_t)row * ldc + col] = v;
    }
  }
}

struct ChunkNN { v2f b[4], a0[4], a1[4], a2[4], a3[4]; };

__device__ __forceinline__ void load_nn(ChunkNN& c, const float* A0,
                                        const float* A1, const float* A2,
                                        const float* A3, const float* Bbase,
                                        int ldb, int k0) {
#pragma unroll
  for (int kk = 0; kk < 4; ++kk) {
    const size_t kb = (size_t)(k0 + 4 * kk) * ldb;
    c.b[kk].x = Bbase[kb];
    c.b[kk].y = Bbase[kb + ldb];
  }
#pragma unroll
  for (int kk = 0; kk < 4; ++kk) c.a0[kk] = *(const v2f*)(A0 + k0 + 4 * kk);
#pragma unroll
  for (int kk = 0; kk < 4; ++kk) c.a1[kk] = *(const v2f*)(A1 + k0 + 4 * kk);
#pragma unroll
  for (int kk = 0; kk < 4; ++kk) c.a2[kk] = *(const v2f*)(A2 + k0 + 4 * kk);
#pragma unroll
  for (int kk = 0; kk < 4; ++kk) c.a3[kk] = *(const v2f*)(A3 + k0 + 4 * kk);
}

__device__ __forceinline__ void mma_nn(v8f acc[4], const ChunkNN& c) {
#pragma unroll
  for (int kk = 0; kk < 4; ++kk) {
    acc[0] = wmma4f32(c.a0[kk], c.b[kk], acc[0]);
    acc[1] = wmma4f32(c.a1[kk], c.b[kk], acc[1]);
    acc[2] = wmma4f32(c.a2[kk], c.b[kk], acc[2]);
    acc[3] = wmma4f32(c.a3[kk], c.b[kk], acc[3]);
  }
}

// C[m,n] = sum_k A[m,k] * Bm[k*ldb + n]   (B not transposed; used for P @ V)
// One wave computes a 64 x 16 block: strided B fragment shared by 4 WMMAs.
// Requires M % 64 == 0, N % 16 == 0, K % 32 == 0.
__global__ __launch_bounds__(WPB * 32)
void gemm_nn_kernel(const float* __restrict__ A, const float* __restrict__ Bm,
                    float* __restrict__ C,
                    int M, int N, int K, int lda, int ldb, int ldc, int nh,
                    size_t aBS, size_t aHS, size_t bBS, size_t bHS,
                    size_t cBS, size_t cHS) {
  const int lane = threadIdx.x & 31;
  const int wave = threadIdx.x >> 5;
  const int tilesN = N >> 4;
  const int tilesM = M >> 6;                    // 64-tall M blocks
  const int tile = blockIdx.x * WPB + wave;
  if (tile >= tilesM * tilesN) return;
  const int tm = tile / tilesN;
  const int tn = tile - tm * tilesN;
  const int r  = lane & 15;
  const int hi = lane >> 4;
  const int zb = blockIdx.z / nh;
  const int zh = blockIdx.z - zb * nh;

  const float* Abase = A + (size_t)zb * aBS + (size_t)zh * aHS + 2 * hi;
  const float* A0 = Abase + (size_t)(tm * 64 + r) * lda;
  const float* A1 = A0 + (size_t)16 * lda;
  const float* A2 = A0 + (size_t)32 * lda;
  const float* A3 = A0 + (size_t)48 * lda;
  const float* Bbase = Bm + (size_t)zb * bBS + (size_t)zh * bHS +
                       (tn * 16 + r) + (size_t)(2 * hi) * ldb;

  v8f acc[4];
#pragma unroll
  for (int j = 0; j < 4; ++j)
    acc[j] = (v8f){0.f, 0.f, 0.f, 0.f, 0.f, 0.f, 0.f, 0.f};

  ChunkNN U, V;
  load_nn(U, A0, A1, A2, A3, Bbase, ldb, 0);
#pragma unroll 1
  for (int k0 = 0; k0 < K; k0 += 32) {
    load_nn(V, A0, A1, A2, A3, Bbase, ldb, k0 + 16);
    mma_nn(acc, U);
    if (k0 + 32 < K)
      load_nn(U, A0, A1, A2, A3, Bbase, ldb, k0 + 32);
    mma_nn(acc, V);
  }

  float* Cb = C + (size_t)zb * cBS + (size_t)zh * cHS;
  const int col = tn * 16 + r;
#pragma unroll
  for (int j = 0; j < 4; ++j) {
#pragma unroll
    for (int i = 0; i < 8; ++i) {
      const int row = tm * 64 + j * 16 + i + 8 * hi;
      Cb[(size_t)row * ldc + col] = acc[j][i];
    }
  }
}

// Faithful mask + softmax over k (row length SEQ).  blockIdx.x = (b*H+h)*S + q.
// mode 0: masked (replace with -1e-9) where tok[b,k] != 0          (src masks)
// mode 1: masked where tok[b,q] != 0 && k <= q                     (tgt mask)
__global__ __launch_bounds__(256)
void mask_softmax_kernel(float* __restrict__ s, const int* __restrict__ tok,
                         int mode) {
  __shared__ float red[256];
  const int row = blockIdx.x;
  const int q  = row & (SEQ - 1);
  const int bh = row >> 9;            // / SEQ
  const int b  = bh >> 3;             // / NHEAD
  float* sp = s + (size_t)row * SEQ;
  const int t = threadIdx.x;
  const int k0 = t, k1 = t + 256;

  float v0 = sp[k0], v1 = sp[k1];
  bool m0, m1;
  if (mode == 0) {
    m0 = tok[b * SEQ + k0] != 0;
    m1 = tok[b * SEQ + k1] != 0;
  } else {
    const bool qa = tok[b * SEQ + q] != 0;
    m0 = qa && (k0 <= q);
    m1 = qa && (k1 <= q);
  }
  if (m0) v0 = -1e-9f;
  if (m1) v1 = -1e-9f;

  red[t] = fmaxf(v0, v1);
  __syncthreads();
  for (int off = 128; off > 0; off >>= 1) {
    if (t < off) red[t] = fmaxf(red[t], red[t + off]);
    __syncthreads();
  }
  const float mx = red[0];
  __syncthreads();

  const float e0 = expf(v0 - mx), e1 = expf(v1 - mx);
  red[t] = e0 + e1;
  __syncthreads();
  for (int off = 128; off > 0; off >>= 1) {
    if (t < off) red[t] += red[t + off];
    __syncthreads();
  }
  const float inv = 1.0f / red[0];
  sp[k0] = e0 * inv;
  sp[k1] = e1 * inv;
}

// x[row] = LayerNorm(x[row] + rsd[row]) * g + b   (in place; D_MODEL = 512)
__global__ __launch_bounds__(256)
void resid_ln_kernel(float* __restrict__ x, const float* __restrict__ rsd,
                     const float* __restrict__ g, const float* __restrict__ bta) {
  __shared__ float red[256];
  __shared__ float stats[2];
  const int row = blockIdx.x;
  const int t = threadIdx.x;
  float* xp = x + (size_t)row * D_MODEL;
  const float* rp = rsd + (size_t)row * D_MODEL;

  const float a0 = xp[t] + rp[t];
  const float a1 = xp[t + 256] + rp[t + 256];

  red[t] = a0 + a1;
  __syncthreads();
  for (int off = 128; off > 0; off >>= 1) {
    if (t < off) red[t] += red[t + off];
    __syncthreads();
  }
  if (t == 0) stats[0] = red[0] * (1.0f / D_MODEL);
  __syncthreads();
  const float m = stats[0];
  const float d0 = a0 - m, d1 = a1 - m;

  red[t] = d0 * d0 + d1 * d1;
  __syncthreads();
  for (int off = 128; off > 0; off >>= 1) {
    if (t < off) red[t] += red[t + off];
    __syncthreads();
  }
  if (t == 0) stats[1] = rsqrtf(red[0] * (1.0f / D_MODEL) + 1e-5f);
  __syncthreads();
  const float rinv = stats[1];

  xp[t]       = d0 * rinv * g[t] + bta[t];
  xp[t + 256] = d1 * rinv * g[t + 256] + bta[t + 256];
}

// out[b,s,:] = emb[tok[b,s],:] * sqrt(D) + pe[s,:]
__global__ __launch_bounds__(256)
void embed_kernel(const int* __restrict__ tok, const float* __restrict__ emb,
                  const float* __restrict__ pe, float* __restrict__ out) {
  const int i = blockIdx.x * 256 + threadIdx.x;   // over B*S*D
  const int d = i & (D_MODEL - 1);
  const int bs = i >> 9;
  const int spos = bs & (SEQ - 1);
  const int tkn = tok[bs];
  out[i] = emb[(size_t)tkn * D_MODEL + d] * EMB_SCALE + pe[spos * D_MODEL + d];
}

// ---------------------------------------------------------------------------

namespace {
struct Lin { const float *w, *b; };
struct Nrm { const float *g, *b; };
struct ATT { Lin q, k, v, o; };
struct EncL { ATT sa; Lin f1, f2; Nrm n1, n2; };
struct DecL { ATT sa, ca; Lin f1, f2; Nrm n1, n2, n3; };

inline void gemm_tn(hipStream_t st, const float* A, const float* W,
                    const float* bias, float* C, int M, int N, int K,
                    int lda, int ldw, int ldc, int nz, int nh,
                    size_t aBS, size_t aHS, size_t wBS, size_t wHS,
                    size_t cBS, size_t cHS, float scale, int act) {
  const int tiles = (M / 16) * (N / 64);
  dim3 grid((tiles + WPB - 1) / WPB, 1, nz);
  gemm_tn_kernel<<<grid, WPB * 32, 0, st>>>(A, W, bias, C, M, N, K, lda, ldw,
                                            ldc, nh, aBS, aHS, wBS, wHS, cBS,
                                            cHS, scale, act);
}

inline void gemm_nn(hipStream_t st, const float* A, const float* Bm, float* C,
                    int M, int N, int K, int lda, int ldb, int ldc, int nz,
                    int nh, size_t aBS, size_t aHS, size_t bBS, size_t bHS,
                    size_t cBS, size_t cHS) {
  const int tiles = (M / 64) * (N / 16);
  dim3 grid((tiles + WPB - 1) / WPB, 1, nz);
  gemm_nn_kernel<<<grid, WPB * 32, 0, st>>>(A, Bm, C, M, N, K, lda, ldb, ldc,
                                            nh, aBS, aHS, bBS, bHS, cBS, cHS);
}
}  // namespace

extern "C" void kernel_launch(void* const* d_in, const int* in_sizes, int n_in,
                              void* d_out, int out_size, void* d_ws,
                              size_t ws_size, hipStream_t stream) {
  (void)in_sizes; (void)n_in; (void)out_size; (void)ws_size;

  // ---- parse inputs in setup_inputs() recursive dict-insertion order ----
  int idx = 0;
  const int* src = (const int*)d_in[idx++];
  const int* tgt = (const int*)d_in[idx++];
  const float* enc_emb = (const float*)d_in[idx++];
  const float* dec_emb = (const float*)d_in[idx++];

  auto getL = [&]() { Lin l; l.w = (const float*)d_in[idx++];
                      l.b = (const float*)d_in[idx++]; return l; };
  auto getN = [&]() { Nrm n; n.g = (const float*)d_in[idx++];
                      n.b = (const float*)d_in[idx++]; return n; };
  auto getA = [&]() { ATT a; a.q = getL(); a.k = getL(); a.v = getL();
                      a.o = getL(); return a; };

  EncL enc[NLAYER];
  DecL dec[NLAYER];
  for (int l = 0; l < NLAYER; ++l) {
    enc[l].sa = getA(); enc[l].f1 = getL(); enc[l].f2 = getL();
    enc[l].n1 = getN(); enc[l].n2 = getN();
  }
  for (int l = 0; l < NLAYER; ++l) {
    dec[l].sa = getA(); dec[l].ca = getA();
    dec[l].f1 = getL(); dec[l].f2 = getL();
    dec[l].n1 = getN(); dec[l].n2 = getN(); dec[l].n3 = getN();
  }
  Lin fc = getL();
  const float* pe = (const float*)d_in[idx++];

  // ---- workspace layout (floats) ----
  float* ws = (float*)d_ws;
  const size_t P = (size_t)BATCH * SEQ * D_MODEL;   // 1 Mi floats
  float* xe  = ws;             // encoder activations [B,S,D]
  float* xd  = ws + 1 * P;     // decoder activations
  float* qb  = ws + 2 * P;
  float* kb  = ws + 3 * P;
  float* vb  = ws + 4 * P;
  float* ab  = ws + 5 * P;     // attention output (heads concat)
  float* tb  = ws + 6 * P;     // sublayer temp
  float* ffb = ws + 7 * P;     // [B,S,FF] = 4P
  float* scb = ws + 11 * P;    // scores [B,H,S,S] = 8P

  const int M = BATCH * SEQ;   // 2048 rows

  auto lin = [&](const float* A, const Lin& L, float* C, int N, int K,
                 int act) {
    gemm_tn(stream, A, L.w, L.b, C, M, N, K, K, K, N, 1, 1,
            0, 0, 0, 0, 0, 0, 1.0f, act);
  };

  // full attention sublayer: out = Wo( softmax(mask(QK^T/8)) V ) + bo
  auto attention = [&](const float* xq, const float* xkv, const ATT& a,
                       const int* mtok, int mode, float* out) {
    lin(xq,  a.q, qb, D_MODEL, D_MODEL, 0);
    lin(xkv, a.k, kb, D_MODEL, D_MODEL, 0);
    lin(xkv, a.v, vb, D_MODEL, D_MODEL, 0);
    // scores[b,h,q,k] = (1/8) * Q[b,q,h*64:] . K[b,k,h*64:]
    gemm_tn(stream, qb, kb, nullptr, scb, SEQ, SEQ, DHEAD,
            D_MODEL, D_MODEL, SEQ, BATCH * NHEAD, NHEAD,
            (size_t)SEQ * D_MODEL, DHEAD,
            (size_t)SEQ * D_MODEL, DHEAD,
            (size_t)NHEAD * SEQ * SEQ, (size_t)SEQ * SEQ, 0.125f, 0);
    mask_softmax_kernel<<<BATCH * NHEAD * SEQ, 256, 0, stream>>>(scb, mtok,
                                                                 mode);
    // ab[b,q,h*64+d] = sum_k P[b,h,q,k] * V[b,k,h*64+d]
    gemm_nn(stream, scb, vb, ab, SEQ, DHEAD, SEQ, SEQ, D_MODEL, D_MODEL,
            BATCH * NHEAD, NHEAD,
            (size_t)NHEAD * SEQ * SEQ, (size_t)SEQ * SEQ,
            (size_t)SEQ * D_MODEL, DHEAD,
            (size_t)SEQ * D_MODEL, DHEAD);
    lin(ab, a.o, out, D_MODEL, D_MODEL, 0);
  };

  // ---------------- encoder ----------------
  embed_kernel<<<(int)(P / 256), 256, 0, stream>>>(src, enc_emb, pe, xe);
  for (int l = 0; l < NLAYER; ++l) {
    attention(xe, xe, enc[l].sa, src, 0, tb);
    resid_ln_kernel<<<M, 256, 0, stream>>>(xe, tb, enc[l].n1.g, enc[l].n1.b);
    lin(xe, enc[l].f1, ffb, FFDIM, D_MODEL, 1);     // + ReLU
    lin(ffb, enc[l].f2, tb, D_MODEL, FFDIM, 0);
    resid_ln_kernel<<<M, 256, 0, stream>>>(xe, tb, enc[l].n2.g, enc[l].n2.b);
  }

  // ---------------- decoder ----------------
  embed_kernel<<<(int)(P / 256), 256, 0, stream>>>(tgt, dec_emb, pe, xd);
  for (int l = 0; l < NLAYER; ++l) {
    attention(xd, xd, dec[l].sa, tgt, 1, tb);       // masked self-attn
    resid_ln_kernel<<<M, 256, 0, stream>>>(xd, tb, dec[l].n1.g, dec[l].n1.b);
    attention(xd, xe, dec[l].ca, src, 0, tb);       // cross-attn vs encoder
    resid_ln_kernel<<<M, 256, 0, stream>>>(xd, tb, dec[l].n2.g, dec[l].n2.b);
    lin(xd, dec[l].f1, ffb, FFDIM, D_MODEL, 1);
    lin(ffb, dec[l].f2, tb, D_MODEL, FFDIM, 0);
    resid_ln_kernel<<<M, 256, 0, stream>>>(xd, tb, dec[l].n3.g, dec[l].n3.b);
  }

  // ---------------- final projection to vocab ----------------
  lin(xd, fc, (float*)d_out, VOCAB, D_MODEL, 0);
}